// Random_Smooth_31026843746600
// MI455X (gfx1250) — compile-verified
//
#include <hip/hip_runtime.h>
#include <stdint.h>

// ---------------- problem constants (match reference) ----------------
#define N_TOT   33554432            // N
#define HALF_M  16777214            // (N-4)/2 : threefry counter split point
#define NPASS   9                   // NUM_ITERATION + 1
#define TILE    2048                // elements per tile (each WG does 2 tiles, HALF_M apart)
#define HALO    18                  // 2 * NPASS
#define EXT     (TILE + 2*HALO)     // 2084 floats per extended tile
#define NTHR    256                 // 8 waves (wave32)
#define NWG     8192                // ceil(HALF_M/TILE): A tiles own [2,HALF+2), B tiles [HALF+2,N)
#define NSLOT   9                   // ceil((EXT-4)/NTHR)

// ---------------- JAX Threefry-2x32-20 (exact), runtime version ----------------
__device__ __forceinline__ uint32_t rotl32(uint32_t x, uint32_t r) {
  return (x << r) | (x >> (32u - r));
}
__device__ __forceinline__ void tf2x32(uint32_t k0, uint32_t k1,
                                       uint32_t c0, uint32_t c1,
                                       uint32_t &o0, uint32_t &o1) {
  const uint32_t ks2 = k0 ^ k1 ^ 0x1BD11BDAu;
  uint32_t x0 = c0 + k0, x1 = c1 + k1;
#define TF_R(r) { x0 += x1; x1 = rotl32(x1, r); x1 ^= x0; }
  TF_R(13) TF_R(15) TF_R(26) TF_R(6)  x0 += k1;  x1 += ks2 + 1u;
  TF_R(17) TF_R(29) TF_R(16) TF_R(24) x0 += ks2; x1 += k0  + 2u;
  TF_R(13) TF_R(15) TF_R(26) TF_R(6)  x0 += k0;  x1 += k1  + 3u;
  TF_R(17) TF_R(29) TF_R(16) TF_R(24) x0 += k1;  x1 += ks2 + 4u;
  TF_R(13) TF_R(15) TF_R(26) TF_R(6)  x0 += ks2; x1 += k0  + 5u;
#undef TF_R
  o0 = x0; o1 = x1;
}

// ---------------- compile-time pass keys: fold_in(key(42), p) ----------------
struct TFK { uint32_t a, b; };
constexpr uint32_t crotl(uint32_t x, int r) { return (x << r) | (x >> (32 - r)); }
constexpr TFK tfkey(uint32_t p) {
  const uint32_t k0 = 0u, k1 = 42u;
  const uint32_t ks[3] = { k0, k1, k0 ^ k1 ^ 0x1BD11BDAu };
  const int rots[2][4] = { {13,15,26,6}, {17,29,16,24} };
  uint32_t x0 = 0u + k0, x1 = p + k1;
  for (int i = 0; i < 5; ++i) {
    for (int j = 0; j < 4; ++j) { x0 += x1; x1 = crotl(x1, rots[i & 1][j]); x1 ^= x0; }
    x0 += ks[(i + 1) % 3];
    x1 += ks[(i + 2) % 3] + (uint32_t)(i + 1);
  }
  return TFK{ x0, x1 };
}
__device__ constexpr TFK PASS_KEYS[NPASS] = {
  tfkey(0), tfkey(1), tfkey(2), tfkey(3), tfkey(4),
  tfkey(5), tfkey(6), tfkey(7), tfkey(8)
};

// ---------------- CDNA5 async DMA helpers (ASYNCcnt-tracked) ----------------
__device__ __forceinline__ uint32_t lds_addr(const void* p) {
  return (uint32_t)(uintptr_t)p;   // low 32 bits of flat LDS pointer = wave-relative LDS byte addr
}
__device__ __forceinline__ void async_load_b32(uint32_t lds, uint32_t voff, const void* base) {
  asm volatile("global_load_async_to_lds_b32 %0, %1, %2"
               :: "v"(lds), "v"(voff), "s"(base) : "memory");
}
__device__ __forceinline__ void async_store_b32(uint32_t voff, uint32_t lds, void* base) {
  asm volatile("global_store_async_from_lds_b32 %0, %1, %2"
               :: "v"(voff), "v"(lds), "s"(base) : "memory");
}
__device__ __forceinline__ void wait_async0() {
  asm volatile("s_wait_asynccnt 0x0" ::: "memory");
}

// ---------------- the 9 fused passes; EDGE instantiation only for blocks 0 / NWG-1 ----------
template<bool EDGE>
__device__ __forceinline__ int run_all_passes(const int t, const int ga0,
                                              float (&sA)[2][EXT], float (&sB)[2][EXT],
                                              const float (&yAv)[NSLOT],
                                              const float (&yBv)[NSLOT]) {
  int cur = 0;
  for (int p = 0; p < NPASS; ++p) {
    const uint32_t kk0 = PASS_KEYS[p].a, kk1 = PASS_KEYS[p].b;
    const int dst = cur ^ 1;
    const float* RA = sA[cur]; float* WA = sA[dst];
    const float* RB = sB[cur]; float* WB = sB[dst];
#pragma unroll
    for (int s = 0; s < NSLOT; ++s) {
      const int j = 2 + t + NTHR * s;            // compute positions [2, EXT-2)
      if (j < EXT - 2) {
        const int ga = ga0 - HALO + j;           // global index, tile A
        const int ma = ga - 2;                   // mask index, tile A
        // paired cipher: ctr (ma, ma+HALF) -> x0 = mask[ma], x1 = mask[ma+HALF]
        uint32_t r0, r1;
        tf2x32(kk0, kk1, (uint32_t)ma, (uint32_t)(ma + HALF_M), r0, r1);
        bool maskA = (r0 >> 31) == 0u;           // uniform<0.5 <=> MSB(bits)==0
        bool maskB = (r1 >> 31) == 0u;
        if (EDGE) {
          if (ma >= HALF_M) {                    // last block's right seam: x1 of (ma-HALF, ma)
            uint32_t q0, q1;
            tf2x32(kk0, kk1, (uint32_t)(ma - HALF_M), (uint32_t)ma, q0, q1);
            maskA = (q1 >> 31) == 0u;
          }
          if (ma < 0) {                          // block 0's B-left seam: x0 of (mb, mb+HALF)
            const int mb = ma + HALF_M;
            uint32_t q0, q1;
            tf2x32(kk0, kk1, (uint32_t)mb, (uint32_t)(mb + HALF_M), q0, q1);
            maskB = (q0 >> 31) == 0u;
          }
        }
        // ---- tile A ----
        {
          const float c = RA[j];
          float st   = RA[j-2] - 4.f*RA[j-1] + 6.f*c - 4.f*RA[j+1] + RA[j+2];
          float term = 0.5f*st + 0.5f*(c - yAv[s]);
          float o = maskA ? c + 0.05f*term : c - 0.1f*term;   // NI_ROUGH / NI_SMOOTH
          if (EDGE) {
            if (!(ga >= 2 && ga <= N_TOT - 3)) {
              if (ga == 1)
                o = c + 0.1f*(0.5f*(-2.f*RA[j-1] + 5.f*c - 4.f*RA[j+1] + RA[j+2])
                              + 0.5f*(c - yAv[s]));
              else
                o = c;                            // ga==0 frozen; ga<0 don't-care
            }
          }
          WA[j] = o;
        }
        // ---- tile B ----
        {
          const float c = RB[j];
          float st   = RB[j-2] - 4.f*RB[j-1] + 6.f*c - 4.f*RB[j+1] + RB[j+2];
          float term = 0.5f*st + 0.5f*(c - yBv[s]);
          float o = maskB ? c + 0.05f*term : c - 0.1f*term;
          if (EDGE) {
            const int gb = ga + HALF_M;
            if (!(gb >= 2 && gb <= N_TOT - 3)) {
              if (gb == N_TOT - 2 || gb == N_TOT - 1) o = yBv[s];  // reset to y each pass
              else o = c;                                          // gb>=N don't-care
            }
          }
          WB[j] = o;
        }
      }
    }
    __syncthreads();
    cur = dst;
  }
  return cur;
}

__global__ __launch_bounds__(NTHR)
void rs_fused(const float* __restrict__ y, float* __restrict__ out) {
  __shared__ float sA[2][EXT];   // ping-pong, tile A (low-half mask domain)
  __shared__ float sB[2][EXT];   // ping-pong, tile B = A + HALF_M
  __shared__ float sYA[EXT];
  __shared__ float sYB[EXT];     // total LDS: 6*2084*4 = 50016 B

  const int t   = threadIdx.x;
  const int b   = (int)blockIdx.x;
  const int ga0 = b * TILE + 2;
  const int gb0 = ga0 + HALF_M;

  // ---- async DMA: y tiles into LDS (clamped addresses; clamped slots never consumed) ----
  for (int j = t; j < EXT; j += NTHR) {
    int ga = ga0 - HALO + j;
    int gb = gb0 - HALO + j;
    int gac = ga < 0 ? 0 : (ga > N_TOT - 1 ? N_TOT - 1 : ga);
    int gbc = gb > N_TOT - 1 ? N_TOT - 1 : gb;
    async_load_b32(lds_addr(&sYA[j]), (uint32_t)(gac * 4), y);
    async_load_b32(lds_addr(&sYB[j]), (uint32_t)(gbc * 4), y);
  }
  wait_async0();
  __syncthreads();
  // buf starts as y; init BOTH ping-pong buffers so never-rewritten edge slots stay defined
  for (int j = t; j < EXT; j += NTHR) {
    float va = sYA[j], vb = sYB[j];
    sA[0][j] = va; sA[1][j] = va;
    sB[0][j] = vb; sB[1][j] = vb;
  }
  __syncthreads();

  // cache this thread's y values in registers (same LDS slots re-read every pass otherwise)
  float yAv[NSLOT], yBv[NSLOT];
#pragma unroll
  for (int s = 0; s < NSLOT; ++s) {
    int j = 2 + t + NTHR * s;
    if (j > EXT - 1) j = EXT - 1;        // clamped reads for skipped slots (values unused)
    yAv[s] = sYA[j];
    yBv[s] = sYB[j];
  }

  int cur;
  if (b == 0 || b == NWG - 1)
    cur = run_all_passes<true >(t, ga0, sA, sB, yAv, yBv);
  else
    cur = run_all_passes<false>(t, ga0, sA, sB, yAv, yBv);

  // ---- async DMA: LDS -> global, central TILE of each tile ----
  const float* FA = sA[cur];
  const float* FB = sB[cur];
#pragma unroll
  for (int s = 0; s < 8; ++s) {
    const int j  = HALO + t + NTHR * s;  // [18, 2066) == 8*256 exactly
    const int ga = ga0 + (j - HALO);
    const int gb = ga + HALF_M;
    if (ga < HALF_M + 2) {               // A tiles own [2, HALF+2) exactly (trims last block)
      async_store_b32((uint32_t)(ga * 4), lds_addr(&FA[j]), out);
    }
    async_store_b32((uint32_t)(gb * 4), lds_addr(&FB[j]), out);  // B owns [HALF+2, N) exactly
  }
  if (b == 0 && t < 2) {                 // g = 0,1 sit at ext positions 16,17 of block 0's A tile
    async_store_b32((uint32_t)(t * 4), lds_addr(&FA[16 + t]), out);
  }
  wait_async0();
}

extern "C" void kernel_launch(void* const* d_in, const int* in_sizes, int n_in,
                              void* d_out, int out_size, void* d_ws, size_t ws_size,
                              hipStream_t stream) {
  (void)in_sizes; (void)n_in; (void)d_ws; (void)ws_size; (void)out_size;
  const float* y = (const float*)d_in[0];
  float* out     = (float*)d_out;
  rs_fused<<<dim3(NWG), dim3(NTHR), 0, stream>>>(y, out);
}